// ASSA_TransformerBlock_38482906972649
// MI455X (gfx1250) — compile-verified
//
#include <hip/hip_runtime.h>
#include <hip/hip_bf16.h>
#include <math.h>

// ---------------------------------------------------------------------------
// MI455X (gfx1250) implementation of the ASSA transformer block.
// All matmuls use V_WMMA_F32_16X16X4_F32 (fp32 in/out, wave32).
// ---------------------------------------------------------------------------

typedef float v2f __attribute__((ext_vector_type(2)));
typedef float v8f __attribute__((ext_vector_type(8)));

__device__ __forceinline__ v8f wmma_f32(v2f a, v2f b, v8f c) {
  // 8 args: (neg_a, A, neg_b, B, c_mod, C, reuse_a, reuse_b)
  return __builtin_amdgcn_wmma_f32_16x16x4_f32(false, a, false, b, (short)0, c,
                                               false, false);
}

// ---------------------------------------------------------------------------
// Generic fp32 WMMA GEMM:  C[M,N] = A[M,K] * Bt[N,K]^T  (+bias[n]) (+resid)
// Requires: M % 64 == 0, N % 128 == 0, K % 4 == 0.
// Block = 256 threads = 8 waves; wave tile 32x32; block tile 64x128.
// ---------------------------------------------------------------------------
__global__ __launch_bounds__(256) void gemm_wmma_f32(
    const float* __restrict__ A, const float* __restrict__ Bt,
    const float* __restrict__ bias, const float* __restrict__ resid,
    float* __restrict__ C, int M, int N, int K) {
  const int lane = threadIdx.x & 31;
  const int wave = threadIdx.x >> 5;
  const int lo = lane & 15;
  const int g  = lane >> 4;

  const int m0 = blockIdx.y * 64 + (wave >> 2) * 32;
  const int n0 = blockIdx.x * 128 + (wave & 3) * 32;

  v8f acc[2][2] = {};

  const float* a0p = A + (size_t)(m0 + lo) * K + 2 * g;
  const float* a1p = A + (size_t)(m0 + 16 + lo) * K + 2 * g;
  const float* b0p = Bt + (size_t)(n0 + lo) * K + 2 * g;
  const float* b1p = Bt + (size_t)(n0 + 16 + lo) * K + 2 * g;

  for (int k0 = 0; k0 < K; k0 += 4) {
    v2f a0 = *(const v2f*)(a0p + k0);
    v2f a1 = *(const v2f*)(a1p + k0);
    v2f b0 = *(const v2f*)(b0p + k0);
    v2f b1 = *(const v2f*)(b1p + k0);
    acc[0][0] = wmma_f32(a0, b0, acc[0][0]);
    acc[0][1] = wmma_f32(a0, b1, acc[0][1]);
    acc[1][0] = wmma_f32(a1, b0, acc[1][0]);
    acc[1][1] = wmma_f32(a1, b1, acc[1][1]);
  }

  for (int mt = 0; mt < 2; ++mt) {
    for (int nt = 0; nt < 2; ++nt) {
      const int nn = n0 + nt * 16 + lo;
      const float bv = bias ? bias[nn] : 0.0f;
      for (int r = 0; r < 8; ++r) {
        const int mm = m0 + mt * 16 + r + 8 * g;
        float val = acc[mt][nt][r] + bv;
        if (resid) val += resid[(size_t)mm * N + nn];
        C[(size_t)mm * N + nn] = val;
      }
    }
  }
}

// ---------------------------------------------------------------------------
// LayerNorm over last dim (256). One wave per row, 8 rows per block.
// ---------------------------------------------------------------------------
__global__ __launch_bounds__(256) void ln_rows_256(
    const float* __restrict__ x, const float* __restrict__ gamma,
    const float* __restrict__ beta, float* __restrict__ out) {
  const int lane = threadIdx.x & 31;
  const int wave = threadIdx.x >> 5;
  const int row = blockIdx.x * 8 + wave;
  const float* xr = x + (size_t)row * 256;
  float v[8], s = 0.f, ss = 0.f;
  for (int i = 0; i < 8; ++i) {
    v[i] = xr[lane + 32 * i];
    s += v[i];
    ss += v[i] * v[i];
  }
  for (int m = 16; m >= 1; m >>= 1) {
    s += __shfl_xor(s, m, 32);
    ss += __shfl_xor(ss, m, 32);
  }
  const float mean = s * (1.0f / 256.0f);
  const float var = ss * (1.0f / 256.0f) - mean * mean;
  const float inv = rsqrtf(var + 1e-5f);
  float* orow = out + (size_t)row * 256;
  for (int i = 0; i < 8; ++i) {
    const int c = lane + 32 * i;
    orow[c] = (v[i] - mean) * inv * gamma[c] + beta[c];
  }
}

// ---------------------------------------------------------------------------
// Fused ASSA attention (flash-style, dual branch relu^2 + softmax).
// qkv layout: [b*1024 + n][768], q at +0, k at +256, v at +512 (head h -> h*32).
// grid (8 row-blocks, 32 b*h); block 256 = 8 waves; 16 query rows per wave.
// ---------------------------------------------------------------------------
__global__ __launch_bounds__(256) void attn_assa_wmma(
    const float* __restrict__ qkv, const float* __restrict__ alpha,
    const float* __restrict__ beta, float* __restrict__ O) {
  __shared__ float sm[8 * 320];
  const int lane = threadIdx.x & 31;
  const int wave = threadIdx.x >> 5;
  const int lo = lane & 15;
  const int g  = lane >> 4;

  const int bh = blockIdx.y;
  const int b  = bh >> 3;
  const int h  = bh & 7;
  const int row0 = blockIdx.x * 128 + wave * 16;

  float* Sst  = sm + wave * 320;   // 16x16 staged scores
  float* rmax = Sst + 256;         // running row max (softmax)
  float* cfac = Sst + 272;         // per-tile rescale factor
  float* dsum = Sst + 288;         // softmax denom accumulator
  float* psum = Sst + 304;         // relu^2 denom accumulator

  const float scale = 0.17677669529663687f;  // 1/sqrt(32)
  const size_t base = (size_t)b * 1024 * 768 + (size_t)h * 32;

  // Q fragments for this wave's 16 rows (16x32, A layout, k-step 4)
  v2f aq[8];
  {
    const float* qp = qkv + base + (size_t)(row0 + lo) * 768 + 2 * g;
    for (int i = 0; i < 8; ++i) aq[i] = *(const v2f*)(qp + 4 * i);
  }

  if (lane < 16) {
    rmax[lane] = -1e30f;
    dsum[lane] = 0.0f;
    psum[lane] = 0.0f;
  }
  asm volatile("s_wait_dscnt 0" ::: "memory");

  v8f odn[2] = {};
  v8f osp[2] = {};

  for (int kt = 0; kt < 64; ++kt) {
    const int kb = kt * 16;

    // ---- S tile = q @ k^T (16x16), 8 wmma over hd=32
    v8f s = {};
    {
      const float* kp = qkv + base + 256 + (size_t)(kb + lo) * 768 + 2 * g;
      for (int i = 0; i < 8; ++i) {
        v2f bk = *(const v2f*)(kp + 4 * i);
        s = wmma_f32(aq[i], bk, s);
      }
    }

    // ---- preload V tile fragments (16 keys x 32 dims) in B layout
    v2f vb[2][4];
    for (int i = 0; i < 4; ++i) {
      const float* vp = qkv + base + 512 + (size_t)(kb + 4 * i + 2 * g) * 768;
      for (int nt = 0; nt < 2; ++nt) {
        v2f t = {vp[nt * 16 + lo], vp[768 + nt * 16 + lo]};
        vb[nt][i] = t;
      }
    }

    // ---- stage scaled scores into LDS (C layout -> row-major)
    for (int r = 0; r < 8; ++r) Sst[(r + 8 * g) * 16 + lo] = s[r] * scale;
    asm volatile("s_wait_dscnt 0" ::: "memory");

    // ---- per-row bookkeeping (lanes 0..15 each own one row)
    if (lane < 16) {
      const int m = lane;
      float sv[16];
      float tm = -1e30f;
      for (int j = 0; j < 16; ++j) {
        sv[j] = Sst[m * 16 + j];
        tm = fmaxf(tm, sv[j]);
      }
      const float om = rmax[m];
      const float nm = fmaxf(om, tm);
      const float cf = __expf(om - nm);
      float da = 0.0f, pa = 0.0f;
      for (int j = 0; j < 16; ++j) {
        da += __expf(sv[j] - nm);
        const float t = fmaxf(sv[j], 0.0f);
        pa += t * t;
      }
      rmax[m] = nm;
      cfac[m] = cf;
      dsum[m] = dsum[m] * cf + da;
      psum[m] = psum[m] + pa;
    }
    asm volatile("s_wait_dscnt 0" ::: "memory");

    // ---- rescale softmax accumulator by correction factor
    float cr[8];
    for (int r = 0; r < 8; ++r) cr[r] = cfac[r + 8 * g];
    for (int nt = 0; nt < 2; ++nt)
      for (int r = 0; r < 8; ++r) odn[nt][r] *= cr[r];

    const float rm = rmax[lo];

    // ---- build A fragments (exp branch, relu^2 branch) and accumulate P@V
    for (int i = 0; i < 4; ++i) {
      const float s0 = Sst[lo * 16 + 4 * i + 2 * g];
      const float s1 = Sst[lo * 16 + 4 * i + 2 * g + 1];
      v2f pe = {__expf(s0 - rm), __expf(s1 - rm)};
      const float t0 = fmaxf(s0, 0.0f), t1 = fmaxf(s1, 0.0f);
      v2f pr = {t0 * t0, t1 * t1};
      for (int nt = 0; nt < 2; ++nt) {
        odn[nt] = wmma_f32(pe, vb[nt][i], odn[nt]);
        osp[nt] = wmma_f32(pr, vb[nt][i], osp[nt]);
      }
    }
  }

  asm volatile("s_wait_dscnt 0" ::: "memory");

  // ---- mix branches and write out (B,N,NH*hd)
  const float av = 1.0f / (1.0f + __expf(-alpha[0]));
  const float bv = 1.0f / (1.0f + __expf(-beta[0]));
  const float wsum = av + bv + 1e-8f;
  const float wa = av / wsum, wb = bv / wsum;

  for (int nt = 0; nt < 2; ++nt) {
    for (int r = 0; r < 8; ++r) {
      const int m = r + 8 * g;
      const float dden = dsum[m];
      const float pden = psum[m] + 1e-8f;
      const float val = wa * osp[nt][r] / pden + wb * odn[nt][r] / dden;
      const int rowg = row0 + m;
      O[((size_t)b * 1024 + rowg) * 256 + h * 32 + nt * 16 + lo] = val;
    }
  }
}

// ---------------------------------------------------------------------------
// Posi-map path: 3x3 SAME conv (NCHW, batch 1, 32x32), out channels = 64
// ---------------------------------------------------------------------------
__global__ __launch_bounds__(256) void conv3x3_s1(
    const float* __restrict__ in, const float* __restrict__ w,
    float* __restrict__ out, int Cin) {
  const int idx = blockIdx.x * 256 + threadIdx.x;  // 64*1024
  const int oc = idx >> 10;
  const int pix = idx & 1023;
  const int y = pix >> 5, x = pix & 31;
  float acc = 0.0f;
  const float* wp = w + (size_t)oc * Cin * 9;
  for (int ic = 0; ic < Cin; ++ic) {
    const float* ip = in + (size_t)ic * 1024;
    for (int ky = 0; ky < 3; ++ky) {
      const int yy = y + ky - 1;
      if (yy < 0 || yy > 31) continue;
      for (int kx = 0; kx < 3; ++kx) {
        const int xx = x + kx - 1;
        if (xx < 0 || xx > 31) continue;
        acc += wp[ic * 9 + ky * 3 + kx] * ip[yy * 32 + xx];
      }
    }
  }
  out[idx] = acc;
}

// Full-tensor mean / inv-std over 65536 elements (axes (1,2,3), batch 1).
__global__ __launch_bounds__(256) void stat_reduce_65536(
    const float* __restrict__ in, float* __restrict__ stats) {
  __shared__ float ss[256], sq[256];
  float s = 0.0f, q = 0.0f;
  for (int i = threadIdx.x; i < 65536; i += 256) {
    const float v = in[i];
    s += v;
    q += v * v;
  }
  ss[threadIdx.x] = s;
  sq[threadIdx.x] = q;
  __syncthreads();
  for (int st = 128; st > 0; st >>= 1) {
    if (threadIdx.x < st) {
      ss[threadIdx.x] += ss[threadIdx.x + st];
      sq[threadIdx.x] += sq[threadIdx.x + st];
    }
    __syncthreads();
  }
  if (threadIdx.x == 0) {
    const float mean = ss[0] * (1.0f / 65536.0f);
    const float var = sq[0] * (1.0f / 65536.0f) - mean * mean;
    stats[0] = mean;
    stats[1] = rsqrtf(var + 1e-5f);
  }
}

__global__ __launch_bounds__(256) void ln_relu_elem(
    const float* __restrict__ in, const float* __restrict__ stats,
    const float* __restrict__ gamma, const float* __restrict__ beta,
    float* __restrict__ out) {
  const int i = blockIdx.x * 256 + threadIdx.x;  // 65536
  const float v = (in[i] - stats[0]) * stats[1] * gamma[i] + beta[i];
  out[i] = fmaxf(v, 0.0f);
}

// im2col for the cf conv: pat[pix][c*9+t] (1024 x 576), zero-padded.
__global__ __launch_bounds__(256) void im2col9_64(
    const float* __restrict__ p, float* __restrict__ pat) {
  const int idx = blockIdx.x * 256 + threadIdx.x;  // 1024*576
  const int pix = idx / 576;
  const int k = idx - pix * 576;
  const int c = k / 9;
  const int t = k - c * 9;
  const int y = pix >> 5, x = pix & 31;
  const int yy = y + t / 3 - 1;
  const int xx = x + t % 3 - 1;
  float v = 0.0f;
  if (yy >= 0 && yy < 32 && xx >= 0 && xx < 32) v = p[c * 1024 + yy * 32 + xx];
  pat[idx] = v;
}

// ---------------------------------------------------------------------------
// TVConv + exact GELU gating.
// wgt_t: [pix][18432] (channel = c*9 + t), y: [b*1024+pix][2048],
// z_t:   [b*1024+pix][1024]
// ---------------------------------------------------------------------------
__global__ __launch_bounds__(256) void tvconv_gelu(
    const float* __restrict__ wgt_t, const float* __restrict__ y,
    float* __restrict__ zt) {
  const int b = blockIdx.x >> 10;
  const int pix = blockIdx.x & 1023;
  const int py = pix >> 5, px = pix & 31;
  const float* wrow = wgt_t + (size_t)pix * 18432;

  const float* ynb[9];
  bool valid[9];
  for (int t = 0; t < 9; ++t) {
    const int yy = py + t / 3 - 1;
    const int xx = px + t % 3 - 1;
    valid[t] = (yy >= 0 && yy < 32 && xx >= 0 && xx < 32);
    ynb[t] = y + ((size_t)b * 1024 + (valid[t] ? yy * 32 + xx : 0)) * 2048;
  }

  for (int i = 0; i < 4; ++i) {
    const int c = threadIdx.x + 256 * i;
    float o1 = 0.0f, o2 = 0.0f;
    const float* w1 = wrow + (size_t)c * 9;
    const float* w2 = wrow + (size_t)(c + 1024) * 9;
    for (int t = 0; t < 9; ++t) {
      const float yv1 = valid[t] ? ynb[t][c] : 0.0f;
      const float yv2 = valid[t] ? ynb[t][c + 1024] : 0.0f;
      o1 += w1[t] * yv1;
      o2 += w2[t] * yv2;
    }
    const float ge = 0.5f * o1 * (1.0f + erff(o1 * 0.70710678118654752f));
    zt[((size_t)b * 1024 + pix) * 1024 + c] = ge * o2;
  }
}

// ---------------------------------------------------------------------------
// Host-side orchestration
// ---------------------------------------------------------------------------
extern "C" void kernel_launch(void* const* d_in, const int* in_sizes, int n_in,
                              void* d_out, int out_size, void* d_ws,
                              size_t ws_size, hipStream_t stream) {
  const float* x      = (const float*)d_in[0];
  const float* n1g    = (const float*)d_in[1];
  const float* n1b    = (const float*)d_in[2];
  const float* qkv_w  = (const float*)d_in[3];
  const float* alpha  = (const float*)d_in[4];
  const float* beta   = (const float*)d_in[5];
  const float* proj_w = (const float*)d_in[6];
  const float* proj_b = (const float*)d_in[7];
  const float* pin_w  = (const float*)d_in[8];
  const float* posi   = (const float*)d_in[9];
  const float* c0w    = (const float*)d_in[10];
  const float* l0g    = (const float*)d_in[11];
  const float* l0b    = (const float*)d_in[12];
  const float* c1w    = (const float*)d_in[13];
  const float* l1g    = (const float*)d_in[14];
  const float* l1b    = (const float*)d_in[15];
  const float* c2w    = (const float*)d_in[16];
  const float* l2g    = (const float*)d_in[17];
  const float* l2b    = (const float*)d_in[18];
  const float* cf_w   = (const float*)d_in[19];
  const float* pout_w = (const float*)d_in[20];
  float* out = (float*)d_out;

  float* ws = (float*)d_ws;
  // workspace layout (floats)
  float* h1    = ws;                         // 4096*256   = 1,048,576
  float* qkvb  = h1 + 1048576;               // 4096*768   = 3,145,728
  float* oattn = qkvb + 3145728;             // 4096*256   = 1,048,576
  float* xa    = oattn + 1048576;            // 4096*256   = 1,048,576
  float* ybuf  = xa + 1048576;               // 4096*2048  = 8,388,608
  float* p0    = ybuf + 8388608;             // 64*1024    = 65,536
  float* p1    = p0 + 65536;                 // 64*1024    = 65,536
  float* stats = p1 + 65536;                 // 64 (pad)
  float* pat   = stats + 64;                 // 1024*576   = 589,824
  float* wgt_t = pat + 589824;               // 1024*18432 = 18,874,368
  float* zt    = wgt_t + 18874368;           // 4096*1024  = 4,194,304

  // 1) LayerNorm(x) -> h1
  ln_rows_256<<<512, 256, 0, stream>>>(x, n1g, n1b, h1);

  // 2) QKV projection: h1(4096x256) @ qkv_w^T(256x768) -> qkvb
  gemm_wmma_f32<<<dim3(768 / 128, 4096 / 64), 256, 0, stream>>>(
      h1, qkv_w, nullptr, nullptr, qkvb, 4096, 768, 256);

  // 3) Fused dual-branch attention -> oattn
  attn_assa_wmma<<<dim3(8, 32), 256, 0, stream>>>(qkvb, alpha, beta, oattn);

  // 4) Output proj + bias + residual: xa = x + oattn @ proj_w^T + proj_b
  gemm_wmma_f32<<<dim3(256 / 128, 4096 / 64), 256, 0, stream>>>(
      oattn, proj_w, proj_b, x, xa, 4096, 256, 256);

  // 5) project_in: y = xa(4096x256) @ pin_w^T(256x2048) -> ybuf [bn][2048]
  gemm_wmma_f32<<<dim3(2048 / 128, 4096 / 64), 256, 0, stream>>>(
      xa, pin_w, nullptr, nullptr, ybuf, 4096, 2048, 256);

  // 6) posi-map conv stack (batch-independent)
  conv3x3_s1<<<256, 256, 0, stream>>>(posi, c0w, p0, 4);
  stat_reduce_65536<<<1, 256, 0, stream>>>(p0, stats);
  ln_relu_elem<<<256, 256, 0, stream>>>(p0, stats, l0g, l0b, p1);

  conv3x3_s1<<<256, 256, 0, stream>>>(p1, c1w, p0, 64);
  stat_reduce_65536<<<1, 256, 0, stream>>>(p0, stats);
  ln_relu_elem<<<256, 256, 0, stream>>>(p0, stats, l1g, l1b, p1);

  conv3x3_s1<<<256, 256, 0, stream>>>(p1, c2w, p0, 64);
  stat_reduce_65536<<<1, 256, 0, stream>>>(p0, stats);
  ln_relu_elem<<<256, 256, 0, stream>>>(p0, stats, l2g, l2b, p1);

  // 7) cf conv as implicit GEMM: pat(1024x576) @ cf_w^T(576x18432) -> wgt_t
  im2col9_64<<<2304, 256, 0, stream>>>(p1, pat);
  gemm_wmma_f32<<<dim3(18432 / 128, 1024 / 64), 256, 0, stream>>>(
      pat, cf_w, nullptr, nullptr, wgt_t, 1024, 18432, 576);

  // 8) TVConv + GELU gating -> zt [bn][1024]
  tvconv_gelu<<<4096, 256, 0, stream>>>(wgt_t, ybuf, zt);

  // 9) project_out + residual: out = xa + zt(4096x1024) @ pout_w^T(1024x256)
  gemm_wmma_f32<<<dim3(256 / 128, 4096 / 64), 256, 0, stream>>>(
      zt, pout_w, nullptr, xa, out, 4096, 256, 1024);
}